// GATv2Backbone_87814901334495
// MI455X (gfx1250) — compile-verified
//
#include <hip/hip_runtime.h>
#include <hip/hip_bf16.h>
#include <math.h>

// ---------------- problem constants (from reference) ----------------
#define N_NODES 50000
#define E_EDGES 800000
#define ETOT    (E_EDGES + N_NODES)   // self loops appended
#define HEADS   4
#define CHP     32                    // channels per head
#define HC      128                   // hidden dim
#define NGRAPH  64
#define SLOPE   0.2f
#define LN_EPS  1e-5f

typedef float v2f __attribute__((ext_vector_type(2)));
typedef float v8f __attribute__((ext_vector_type(8)));

// ---------------- small helpers ----------------
__device__ __forceinline__ unsigned f32_to_ord(float f) {
  unsigned u = __float_as_uint(f);
  return (u & 0x80000000u) ? ~u : (u | 0x80000000u);
}
__device__ __forceinline__ float ord_to_f32(unsigned u) {
  u = (u & 0x80000000u) ? (u & 0x7fffffffu) : ~u;
  return __uint_as_float(u);
}
__device__ __forceinline__ void atomAddF(float* p, float v) {
  unsafeAtomicAdd(p, v);   // lowers to global_atomic_add_f32 on gfx1250
}

__global__ void fill_f32(float* __restrict__ p, float v, int n) {
  int i = blockIdx.x * blockDim.x + threadIdx.x;
  if (i < n) p[i] = v;
}
__global__ void fill_u32(unsigned* __restrict__ p, unsigned v, int n) {
  int i = blockIdx.x * blockDim.x + threadIdx.x;
  if (i < n) p[i] = v;
}

// ---------------- 1) Y = X @ W + b via V_WMMA_F32_16X16X4_F32 ----------------
// Block: 256 threads (8 waves). W (128x128, row-major W[k][n]) is staged once
// into LDS (stride 136 floats -> disjoint bank sets for the two half-waves),
// then the block sweeps ROWTILES row tiles against it: 64 output rows per
// block, amortizing the 64KB W stage 4x. Wave w owns columns [16w, 16w+16).
#define LDSW 136
#define ROWTILES 4
__global__ void gemm_bias_wmma(const float* __restrict__ X,
                               const float* __restrict__ W,
                               const float* __restrict__ bias,
                               float* __restrict__ Y, int n) {
  extern __shared__ float sW[];            // [128][LDSW]
  const int tid = threadIdx.x;
  // cooperative W load: 4096 float4 / 256 threads = 16 each
  for (int i = 0; i < 16; ++i) {
    int idx = tid + 256 * i;               // float4 index
    int r = idx >> 5, q = idx & 31;
    float4 w4 = *(const float4*)(W + r * HC + q * 4);
    float* d = sW + r * LDSW + q * 4;
    d[0] = w4.x; d[1] = w4.y; d[2] = w4.z; d[3] = w4.w;
  }
  __syncthreads();

  const int wave = tid >> 5;
  const int lane = tid & 31;
  const int m    = lane & 15;              // A row / B col within tile
  const int hi   = lane >> 4;              // K half select (ISA 16x4 layout)
  const int col  = wave * 16 + m;
  const float bc = bias[col];
  const int rowbase = blockIdx.x * (16 * ROWTILES);

  for (int t = 0; t < ROWTILES; ++t) {
    int row0 = rowbase + t * 16;
    if (row0 >= n) break;                  // block-uniform: EXEC stays all-1s
    int row  = row0 + m;
    int rowc = (row < n) ? row : (n - 1);  // clamp reads, predicate stores
    // A: lane holds X[row][k0 + 2*hi + {0,1}]  (ISA 32-bit A 16x4 layout)
    const float* ap = X + (size_t)rowc * HC + 2 * hi;
    v8f acc = {};
    #pragma unroll
    for (int k0 = 0; k0 < HC; k0 += 4) {
      v2f a; a.x = ap[k0]; a.y = ap[k0 + 1];
      // B: lane holds W[k0 + 2*hi + {0,1}][col]  (4x16 B, K split at halves)
      const float* bp = sW + (k0 + 2 * hi) * LDSW + col;
      v2f b; b.x = bp[0]; b.y = bp[LDSW];
      acc = __builtin_amdgcn_wmma_f32_16x16x4_f32(
          false, a, false, b, (short)0, acc, false, false);
    }
    #pragma unroll
    for (int r = 0; r < 8; ++r) {          // C/D layout: VGPR r -> M = r + 8*hi
      int orow = row0 + r + 8 * hi;
      if (orow < n) Y[(size_t)orow * HC + col] = acc[r] + bc;
    }
  }
}

// ---------------- 2) per-edge GATv2 score + segment max ----------------
// one wave per edge; lane l covers channels {l, l+32, l+64, l+96} = one per head
__global__ void edge_score_kernel(const int* __restrict__ ei,
                                  const float* __restrict__ xl,
                                  const float* __restrict__ xr,
                                  const float* __restrict__ att,
                                  float* __restrict__ score,
                                  unsigned* __restrict__ mord) {
  int e = blockIdx.x * 8 + (threadIdx.x >> 5);
  int lane = threadIdx.x & 31;
  if (e >= ETOT) return;
  int src, dst;
  if (e < E_EDGES) { src = ei[e]; dst = ei[E_EDGES + e]; }
  else             { src = dst = e - E_EDGES; }
  const float* pl = xl + (size_t)src * HC + lane;
  const float* pr = xr + (size_t)dst * HC + lane;
  float s[HEADS];
  #pragma unroll
  for (int h = 0; h < HEADS; ++h) {
    float v = pl[h * CHP] + pr[h * CHP];
    v = (v > 0.f) ? v : v * SLOPE;
    s[h] = v * att[h * CHP + lane];
  }
  #pragma unroll
  for (int h = 0; h < HEADS; ++h)
    for (int off = 16; off > 0; off >>= 1) s[h] += __shfl_xor(s[h], off, 32);
  if (lane < HEADS) {
    float sh = s[lane];
    score[(size_t)e * HEADS + lane] = sh;
    atomicMax(&mord[(size_t)dst * HEADS + lane], f32_to_ord(sh));
  }
}

// ---------------- 3) exp(score - max) + segment sum ----------------
__global__ void edge_exp_kernel(const int* __restrict__ ei,
                                float* __restrict__ score,      // in: score, out: ex
                                const unsigned* __restrict__ mord,
                                float* __restrict__ denom) {
  int t = blockIdx.x * blockDim.x + threadIdx.x;
  if (t >= ETOT * HEADS) return;
  int e = t >> 2, h = t & 3;
  int dst = (e < E_EDGES) ? ei[E_EDGES + e] : (e - E_EDGES);
  float m = ord_to_f32(mord[(size_t)dst * HEADS + h]);
  float ex = __expf(score[t] - m);
  score[t] = ex;
  atomAddF(&denom[(size_t)dst * HEADS + h], ex);
}

// ---------------- 4) alpha-weighted scatter-add of xl[src] ----------------
__global__ void edge_aggregate_kernel(const int* __restrict__ ei,
                                      const float* __restrict__ xl,
                                      const float* __restrict__ ex,
                                      const float* __restrict__ denom,
                                      float* __restrict__ agg) {
  int e = blockIdx.x * 8 + (threadIdx.x >> 5);
  int lane = threadIdx.x & 31;
  if (e >= ETOT) return;
  int src, dst;
  if (e < E_EDGES) { src = ei[e]; dst = ei[E_EDGES + e]; }
  else             { src = dst = e - E_EDGES; }
  #pragma unroll
  for (int h = 0; h < HEADS; ++h) {
    float alpha = ex[(size_t)e * HEADS + h] / denom[(size_t)dst * HEADS + h];
    float v = xl[(size_t)src * HC + h * CHP + lane] * alpha;
    atomAddF(&agg[(size_t)dst * HC + h * CHP + lane], v);
  }
}

// ---------------- 5) bias + LayerNorm + SiLU (wave per node) ----------------
__global__ void ln_silu_kernel(const float* __restrict__ agg,
                               const float* __restrict__ bias,
                               const float* __restrict__ gamma,
                               const float* __restrict__ beta,
                               float* __restrict__ out) {
  int node = blockIdx.x * 8 + (threadIdx.x >> 5);
  int lane = threadIdx.x & 31;
  if (node >= N_NODES) return;
  float v[4]; float sum = 0.f;
  #pragma unroll
  for (int i = 0; i < 4; ++i) {
    int c = i * 32 + lane;
    v[i] = agg[(size_t)node * HC + c] + bias[c];
    sum += v[i];
  }
  for (int off = 16; off > 0; off >>= 1) sum += __shfl_xor(sum, off, 32);
  float mu = sum * (1.0f / HC);
  float sq = 0.f;
  #pragma unroll
  for (int i = 0; i < 4; ++i) { float d = v[i] - mu; sq += d * d; }
  for (int off = 16; off > 0; off >>= 1) sq += __shfl_xor(sq, off, 32);
  float rstd = rsqrtf(sq * (1.0f / HC) + LN_EPS);
  #pragma unroll
  for (int i = 0; i < 4; ++i) {
    int c = i * 32 + lane;
    float h = (v[i] - mu) * rstd * gamma[c] + beta[c];
    out[(size_t)node * HC + c] = h / (1.0f + __expf(-h));   // SiLU
  }
}

// ---------------- 6) global mean pool ----------------
__global__ void pool_accum_kernel(const float* __restrict__ x,
                                  const int* __restrict__ batch,
                                  float* __restrict__ sums,
                                  float* __restrict__ counts) {
  int node = blockIdx.x * 8 + (threadIdx.x >> 5);
  int lane = threadIdx.x & 31;
  if (node >= N_NODES) return;
  int g = batch[node];
  #pragma unroll
  for (int i = 0; i < 4; ++i) {
    int c = i * 32 + lane;
    atomAddF(&sums[(size_t)g * HC + c], x[(size_t)node * HC + c]);
  }
  if (lane == 0) atomAddF(&counts[g], 1.0f);
}
__global__ void pool_final_kernel(const float* __restrict__ sums,
                                  const float* __restrict__ counts,
                                  float* __restrict__ out) {
  int t = blockIdx.x * blockDim.x + threadIdx.x;
  if (t >= NGRAPH * HC) return;
  out[t] = sums[t] / fmaxf(counts[t >> 7], 1.0f);
}

// ---------------- host orchestration ----------------
extern "C" void kernel_launch(void* const* d_in, const int* in_sizes, int n_in,
                              void* d_out, int out_size, void* d_ws, size_t ws_size,
                              hipStream_t stream) {
  (void)in_sizes; (void)n_in; (void)out_size; (void)ws_size;
  const float* x     = (const float*)d_in[0];
  const int*   ei    = (const int*)d_in[1];
  const int*   batch = (const int*)d_in[2];
  const float* Wl[2]    = {(const float*)d_in[3],  (const float*)d_in[11]};
  const float* Wr[2]    = {(const float*)d_in[4],  (const float*)d_in[12]};
  const float* bl[2]    = {(const float*)d_in[5],  (const float*)d_in[13]};
  const float* br[2]    = {(const float*)d_in[6],  (const float*)d_in[14]};
  const float* att[2]   = {(const float*)d_in[7],  (const float*)d_in[15]};
  const float* bias[2]  = {(const float*)d_in[8],  (const float*)d_in[16]};
  const float* gamma[2] = {(const float*)d_in[9],  (const float*)d_in[17]};
  const float* beta[2]  = {(const float*)d_in[10], (const float*)d_in[18]};

  float* ws = (float*)d_ws;
  float*    xl    = ws;            ws += (size_t)N_NODES * HC;
  float*    xr2   = ws;            ws += (size_t)N_NODES * HC;
  float*    agg   = ws;            ws += (size_t)N_NODES * HC;
  float*    xnode = ws;            ws += (size_t)N_NODES * HC;
  float*    exbuf = ws;            ws += (size_t)ETOT * HEADS;
  float*    denom = ws;            ws += (size_t)N_NODES * HEADS;
  unsigned* mord  = (unsigned*)ws; ws += (size_t)N_NODES * HEADS;
  float*    psums = ws;            ws += (size_t)NGRAPH * HC;
  /* pcnt immediately follows psums (zeroed together) */
  float*    pcnt  = ws;

  dim3 b256(256);
  const int    gemmGrid = (N_NODES + 16 * ROWTILES - 1) / (16 * ROWTILES);
  const size_t shmem    = (size_t)HC * LDSW * sizeof(float);
  const int    edgeGrid = (ETOT + 7) / 8;
  const int    nodeGrid = (N_NODES + 7) / 8;
  const unsigned ORD_NEG_INF = 0x007FFFFFu;   // f32_to_ord(-inf)

  const float* xin = x;
  for (int l = 0; l < 2; ++l) {
    gemm_bias_wmma<<<gemmGrid, b256, shmem, stream>>>(xin, Wl[l], bl[l], xl, N_NODES);
    gemm_bias_wmma<<<gemmGrid, b256, shmem, stream>>>(xin, Wr[l], br[l], xr2, N_NODES);
    fill_u32<<<(N_NODES * HEADS + 255) / 256, b256, 0, stream>>>(mord, ORD_NEG_INF, N_NODES * HEADS);
    fill_f32<<<(N_NODES * HEADS + 255) / 256, b256, 0, stream>>>(denom, 0.f, N_NODES * HEADS);
    fill_f32<<<(N_NODES * HC + 255) / 256, b256, 0, stream>>>(agg, 0.f, N_NODES * HC);
    edge_score_kernel<<<edgeGrid, b256, 0, stream>>>(ei, xl, xr2, att[l], exbuf, mord);
    edge_exp_kernel<<<(ETOT * HEADS + 255) / 256, b256, 0, stream>>>(ei, exbuf, mord, denom);
    edge_aggregate_kernel<<<edgeGrid, b256, 0, stream>>>(ei, xl, exbuf, denom, agg);
    ln_silu_kernel<<<nodeGrid, b256, 0, stream>>>(agg, bias[l], gamma[l], beta[l], xnode);
    xin = xnode;
  }
  fill_f32<<<(NGRAPH * HC + NGRAPH + 255) / 256, b256, 0, stream>>>(psums, 0.f, NGRAPH * HC + NGRAPH);
  pool_accum_kernel<<<nodeGrid, b256, 0, stream>>>(xnode, batch, psums, pcnt);
  pool_final_kernel<<<(NGRAPH * HC + 255) / 256, b256, 0, stream>>>(psums, pcnt, (float*)d_out);
}